// ForwardKinematicsLayer_60696477827277
// MI455X (gfx1250) — compile-verified
//
#include <hip/hip_runtime.h>

typedef __attribute__((ext_vector_type(2))) float v2f;
typedef __attribute__((ext_vector_type(4))) float v4f;
typedef __attribute__((ext_vector_type(8))) float v8f;

#define JNT 24
#define WAVES_PER_WG 4
#define BATCH_PER_WAVE 16
#define BATCH_PER_WG (WAVES_PER_WG * BATCH_PER_WAVE) /* 64 */
#define THREADS (WAVES_PER_WG * 32)                  /* 128 */

// SMPL kinematic tree
__device__ __constant__ int d_par[JNT] = {
    -1, 0, 0, 0, 1, 2, 3, 4, 5, 6, 7, 8, 9, 9, 9, 12, 13, 14, 16, 17, 18, 19, 20, 21};

// ANC[j] bit k == 1 iff joint k lies on the path root->j (incl. j).
// pos[j] = sum_k ANC[j][k] * rotated[k]  (bit 0 contributes rotated[0]==0).
__device__ __constant__ unsigned d_anc[JNT] = {
    0x000001u, 0x000003u, 0x000005u, 0x000009u, 0x000013u, 0x000025u,
    0x000049u, 0x000093u, 0x000125u, 0x000249u, 0x000493u, 0x000925u,
    0x001249u, 0x002249u, 0x004249u, 0x009249u, 0x012249u, 0x024249u,
    0x052249u, 0x0A4249u, 0x152249u, 0x2A4249u, 0x552249u, 0xAA4249u};

__global__ __launch_bounds__(THREADS) void fk_wmma_kernel(
    const v4f* __restrict__ rot4,      // [B, 24] quats (w,x,y,z)
    const float* __restrict__ skel,    // [24, 3]
    float* __restrict__ out,           // [B, 24, 3]
    int nbatch)
{
    __shared__ float lds_off[JNT * 3];
    // per-wave rotated bone vectors: [wave][comp][joint][batch_in_wave]
    __shared__ float lds_rot[WAVES_PER_WG * 3 * JNT * BATCH_PER_WAVE];
    // staged output: [batch_in_wg][joint][comp]  (matches global layout)
    __shared__ float lds_pos[BATCH_PER_WG * JNT * 3];

    const int tid   = threadIdx.x;
    const int w     = tid >> 5;
    const int lane  = tid & 31;
    const int half  = lane >> 4;   // 0: lanes 0-15, 1: lanes 16-31
    const int nlane = lane & 15;
    const size_t wg_base = (size_t)blockIdx.x * BATCH_PER_WG;

    // ---- bone offsets: off[j] = skel[j] - skel[parent(j)] ----
    if (tid < JNT) {
        int p = d_par[tid];
        if (p < 0) p = 0;
        lds_off[tid * 3 + 0] = skel[tid * 3 + 0] - skel[p * 3 + 0];
        lds_off[tid * 3 + 1] = skel[tid * 3 + 1] - skel[p * 3 + 1];
        lds_off[tid * 3 + 2] = skel[tid * 3 + 2] - skel[p * 3 + 2];
    }
    __syncthreads();

    // ---- phase 1: quat -> R -> rotated = R * offset ----
    // Streaming 192 MB of quats (> 192 MB L2, never re-read): NT loads.
    float* rotw = &lds_rot[w * (3 * JNT * BATCH_PER_WAVE)];
    const size_t wave_elem0 = (wg_base + (size_t)w * BATCH_PER_WAVE) * JNT;
#pragma unroll
    for (int i = 0; i < 12; ++i) {          // 16 batches * 24 joints / 32 lanes
        int p  = lane + i * 32;             // 0..383, contiguous across wave
        int bl = p / JNT;                   // batch within wave
        int j  = p - bl * JNT;              // joint
        size_t gb = wg_base + (size_t)w * BATCH_PER_WAVE + bl;
        v4f q = __builtin_nontemporal_load(
            &rot4[(gb < (size_t)nbatch) ? (wave_elem0 + p) : 0]);
        float qw = q.x, qx = q.y, qy = q.z, qz = q.w;
        float ox = lds_off[j * 3 + 0], oy = lds_off[j * 3 + 1], oz = lds_off[j * 3 + 2];
        float r00 = 1.f - 2.f * (qy * qy + qz * qz);
        float r01 = 2.f * (qx * qy - qz * qw);
        float r02 = 2.f * (qx * qz + qy * qw);
        float r10 = 2.f * (qx * qy + qz * qw);
        float r11 = 1.f - 2.f * (qx * qx + qz * qz);
        float r12 = 2.f * (qy * qz - qx * qw);
        float r20 = 2.f * (qx * qz - qy * qw);
        float r21 = 2.f * (qy * qz + qx * qw);
        float r22 = 1.f - 2.f * (qx * qx + qy * qy);
        float rx = r00 * ox + r01 * oy + r02 * oz;
        float ry = r10 * ox + r11 * oy + r12 * oz;
        float rz = r20 * ox + r21 * oy + r22 * oz;
        rotw[(0 * JNT + j) * BATCH_PER_WAVE + bl] = rx;
        rotw[(1 * JNT + j) * BATCH_PER_WAVE + bl] = ry;
        rotw[(2 * JNT + j) * BATCH_PER_WAVE + bl] = rz;
    }
    __syncthreads();

    // ---- phase 2: pos = M(24x24, binary) @ rotated(24 x 16) via WMMA f32 16x16x4 ----
    // A layout: v0 = A[m][kb], v1 = A[m][kb+1]; lanes 0-15 -> K pair (0,1), 16-31 -> (2,3)
    // B layout (mirror):  v0 = B[kb][n], v1 = B[kb+1][n], n = lane&15
    // D layout: VGPR r -> row rt*16 + r + 8*half, col = lane&15
    float* posw = &lds_pos[w * (BATCH_PER_WAVE * JNT * 3)];
#pragma unroll
    for (int rt = 0; rt < 2; ++rt) {
        int mrow = rt * 16 + nlane;
        unsigned mask = (mrow < JNT) ? d_anc[mrow] : 0u;
        // Materialize this lane's 12 A-matrix entries (0/1) once per row tile.
        float aval[12];
#pragma unroll
        for (int kk = 0; kk < 6; ++kk) {
            int kb = kk * 4 + 2 * half;
            aval[kk * 2 + 0] = ((mask >> kb) & 1u) ? 1.0f : 0.0f;
            aval[kk * 2 + 1] = ((mask >> (kb + 1)) & 1u) ? 1.0f : 0.0f;
        }
#pragma unroll
        for (int c = 0; c < 3; ++c) {
            v8f acc = {0.f, 0.f, 0.f, 0.f, 0.f, 0.f, 0.f, 0.f};
#pragma unroll
            for (int kk = 0; kk < 6; ++kk) {     // K = 24 = 6 * 4
                int kb = kk * 4 + 2 * half;
                v2f a, b;
                a.x = aval[kk * 2 + 0];
                a.y = aval[kk * 2 + 1];
                b.x = rotw[(c * JNT + kb) * BATCH_PER_WAVE + nlane];
                b.y = rotw[(c * JNT + kb + 1) * BATCH_PER_WAVE + nlane];
                acc = __builtin_amdgcn_wmma_f32_16x16x4_f32(
                    false, a, false, b, (short)0, acc, false, false);
            }
#pragma unroll
            for (int r = 0; r < 8; ++r) {
                int row = rt * 16 + r + 8 * half;
                if (row < JNT)
                    posw[nlane * (JNT * 3) + row * 3 + c] = acc[r];
            }
        }
    }
    __syncthreads();

    // ---- phase 3: coalesced NT float4 stores of the whole WG tile ----
    // Output (144 MB) is write-once streaming: NT avoids L2 write-allocate thrash.
    const v4f* pos4 = (const v4f*)lds_pos;
    v4f* out4 = (v4f*)(out + wg_base * (JNT * 3));
    const size_t tile_floats = (size_t)nbatch * (JNT * 3) - wg_base * (JNT * 3);
#pragma unroll
    for (int i = 0; i < (BATCH_PER_WG * JNT * 3 / 4) / THREADS; ++i) {  // 9
        int idx = tid + i * THREADS;
        if ((size_t)idx * 4 < tile_floats)
            __builtin_nontemporal_store(pos4[idx], &out4[idx]);
    }
}

extern "C" void kernel_launch(void* const* d_in, const int* in_sizes, int n_in,
                              void* d_out, int out_size, void* d_ws, size_t ws_size,
                              hipStream_t stream) {
    (void)n_in; (void)d_ws; (void)ws_size; (void)out_size;
    const v4f* rot = (const v4f*)d_in[0];        // [B,24,4] f32
    const float* skl = (const float*)d_in[1];    // [24,3]   f32
    float* out = (float*)d_out;                  // [B,24,3] f32
    int B = in_sizes[0] / (JNT * 4);
    int grid = (B + BATCH_PER_WG - 1) / BATCH_PER_WG;   // 8192 for B=524288
    fk_wmma_kernel<<<grid, THREADS, 0, stream>>>(rot, skl, out, B);
}